// ExpertChoiceRouter_35553739276863
// MI455X (gfx1250) — compile-verified
//
#include <hip/hip_runtime.h>

typedef __attribute__((ext_vector_type(2))) float v2f;
typedef __attribute__((ext_vector_type(8))) float v8f;

#define TOKENS  8192
#define HIDDEN  4096
#define EXPERTS 64
#define CAP     160   // int(8192 * 1.25 / 64)

// ---------------------------------------------------------------------------
// Kernel 0: zero the whole output (mask starts at 0; loss is identically 0
// because every expert column of the dispatch mask has exactly CAP ones).
// ---------------------------------------------------------------------------
__global__ void zero_out_kernel(float* __restrict__ out, int n) {
    int i = blockIdx.x * blockDim.x + threadIdx.x;
    if (i < n) out[i] = 0.0f;
}

// ---------------------------------------------------------------------------
// Kernel 1: gate GEMM, fp32 WMMA 16x16x4.
// One wave computes a 16(M) x 64(N) strip => x is read from HBM exactly once.
// A frag (16x4 f32): lane L<16 -> {A[L][k],A[L][k+1]}, lane L+16 -> {A[L][k+2],A[L][k+3]}
// B frag (4x16 f32): VGPR0 = K=khalf row, VGPR1 = K=khalf+1 row (lane = N col)
// Both are a single contiguous float2 load per lane per K-step of 4.
// ---------------------------------------------------------------------------
__global__ __launch_bounds__(256)
void gate_gemm_kernel(const float* __restrict__ x,
                      const float* __restrict__ w,
                      float* __restrict__ logits) {
    const int lane  = threadIdx.x & 31;
    const int wave  = threadIdx.x >> 5;
    const int m0    = (blockIdx.x * 8 + wave) * 16;   // 16 token rows per wave
    const int lrow  = lane & 15;
    const int khalf = (lane >> 4) * 2;                // 0 (lanes 0-15) or 2 (lanes 16-31)

    const float* aptr  = x + (size_t)(m0 + lrow) * HIDDEN + khalf;
    const float* bptr0 = w + (size_t)( 0 + lrow) * HIDDEN + khalf;  // experts 0..15
    const float* bptr1 = w + (size_t)(16 + lrow) * HIDDEN + khalf;  // experts 16..31
    const float* bptr2 = w + (size_t)(32 + lrow) * HIDDEN + khalf;  // experts 32..47
    const float* bptr3 = w + (size_t)(48 + lrow) * HIDDEN + khalf;  // experts 48..63

    v8f c0 = {}, c1 = {}, c2 = {}, c3 = {};

    #pragma unroll 4
    for (int k = 0; k < HIDDEN; k += 4) {
        v2f a  = *(const v2f*)(aptr  + k);
        v2f b0 = *(const v2f*)(bptr0 + k);
        v2f b1 = *(const v2f*)(bptr1 + k);
        v2f b2 = *(const v2f*)(bptr2 + k);
        v2f b3 = *(const v2f*)(bptr3 + k);
        c0 = __builtin_amdgcn_wmma_f32_16x16x4_f32(false, a, false, b0, (short)0, c0, false, false);
        c1 = __builtin_amdgcn_wmma_f32_16x16x4_f32(false, a, false, b1, (short)0, c1, false, false);
        c2 = __builtin_amdgcn_wmma_f32_16x16x4_f32(false, a, false, b2, (short)0, c2, false, false);
        c3 = __builtin_amdgcn_wmma_f32_16x16x4_f32(false, a, false, b3, (short)0, c3, false, false);
    }

    // C/D layout: VGPR v, lanes 0-15 => row m0+v, lanes 16-31 => row m0+v+8; col = lane&15
    const int rbase = m0 + (lane >> 4) * 8;
    const int col   = lane & 15;
    float* outp = logits + (size_t)rbase * EXPERTS + col;
    #pragma unroll
    for (int v = 0; v < 8; ++v) {
        outp[(size_t)v * EXPERTS +  0] = c0[v];
        outp[(size_t)v * EXPERTS + 16] = c1[v];
        outp[(size_t)v * EXPERTS + 32] = c2[v];
        outp[(size_t)v * EXPERTS + 48] = c3[v];
    }
}

// ---------------------------------------------------------------------------
// Kernel 2: per-expert top-CAP (sorted descending, ties -> lower token index,
// matching jax.lax.top_k stability). One block per expert; the 8192-float
// logit column lives in LDS (32 KB of the 320 KB WGP LDS). Deterministic.
// Writes expert_indices (as float) and scatters 1.0 into the dispatch mask.
// ---------------------------------------------------------------------------
__global__ __launch_bounds__(256)
void expert_topk_kernel(const float* __restrict__ logits, float* __restrict__ out) {
    const int e   = blockIdx.x;
    const int tid = threadIdx.x;

    __shared__ float vals[TOKENS];
    __shared__ float rv[256];
    __shared__ int   ri[256];

    for (int i = tid; i < TOKENS; i += 256)
        vals[i] = logits[(size_t)i * EXPERTS + e];
    __syncthreads();

    float* idx_out = out + (size_t)e * CAP;                 // expert_indices[e, :]
    float* mask    = out + (size_t)EXPERTS * CAP;           // dispatch_mask base

    const float NEG_INF = -__builtin_inff();

    for (int c = 0; c < CAP; ++c) {
        // local argmax over strided slice (tie -> smaller index)
        float bv = NEG_INF; int bi = TOKENS;
        for (int i = tid; i < TOKENS; i += 256) {
            float v = vals[i];
            if (v > bv || (v == bv && i < bi)) { bv = v; bi = i; }
        }
        rv[tid] = bv; ri[tid] = bi;
        __syncthreads();
        // tree reduce across 256 threads
        for (int s = 128; s > 0; s >>= 1) {
            if (tid < s) {
                float ov = rv[tid + s]; int oi = ri[tid + s];
                if (ov > rv[tid] || (ov == rv[tid] && oi < ri[tid])) {
                    rv[tid] = ov; ri[tid] = oi;
                }
            }
            __syncthreads();
        }
        if (tid == 0) {
            int wtok = ri[0];
            idx_out[c] = (float)wtok;
            mask[(size_t)wtok * EXPERTS + e] = 1.0f;
            vals[wtok] = NEG_INF;   // remove from future rounds
        }
        __syncthreads();
    }
    // load_balancing_loss: every expert column has exactly CAP ones, so
    // expert_load == mean == CAP and the loss is exactly 0 (written by zeroing).
}

// ---------------------------------------------------------------------------
extern "C" void kernel_launch(void* const* d_in, const int* in_sizes, int n_in,
                              void* d_out, int out_size, void* d_ws, size_t ws_size,
                              hipStream_t stream) {
    const float* x = (const float*)d_in[0];   // [8192, 4096]
    const float* w = (const float*)d_in[1];   // [64, 4096]
    float* out     = (float*)d_out;           // [64*160] idx + [8192*64] mask + [1] loss
    float* logits  = (float*)d_ws;            // [8192, 64] scratch (2 MB)

    // 1) zero outputs (mask + loss; indices are fully overwritten anyway)
    zero_out_kernel<<<(out_size + 255) / 256, 256, 0, stream>>>(out, out_size);

    // 2) gate GEMM: 512 waves (64 blocks x 8 waves), each a 16x64 output strip
    gate_gemm_kernel<<<TOKENS / (16 * 8), 256, 0, stream>>>(x, w, logits);

    // 3) per-expert top-160 + mask scatter
    expert_topk_kernel<<<EXPERTS, 256, 0, stream>>>(logits, out);
}